// MatchGroup_49503793054309
// MI455X (gfx1250) — compile-verified
//
#include <hip/hip_runtime.h>
#include <hip/hip_bf16.h>
#include <math.h>

typedef __attribute__((ext_vector_type(2))) float v2f;
typedef __attribute__((ext_vector_type(8))) float v8f;

#define KS    7
#define TAPS  49      // 7*7
#define NCH   12
#define IMG   512
#define HW    (512*512)
#define TILE_STRIDE 24   // padded LDS row stride (floats)

// ---------------------------------------------------------------------------
// Kernel 1: build 12 rotated 7x7 kernels (bilinear resample on rotated grid)
// and pack them into WMMA A-fragment layout for V_WMMA_F32_16X16X4_F32:
//   wprep[t][lane][0..1] : lane L holds A[M = L%16][K = 4t + 2*(L/16) + {0,1}]
//   (rows 12..15 and K >= 49 are zero padding)
// ---------------------------------------------------------------------------
__global__ __launch_bounds__(640) void rot_prep_kernel(
    const float* __restrict__ k2d, float* __restrict__ wprep)
{
    __shared__ float kk[TAPS];
    __shared__ float rot[NCH * TAPS];
    const int tid = threadIdx.x;

    if (tid < TAPS) kk[tid] = k2d[tid];
    __syncthreads();

    if (tid < NCH * TAPS) {
        const int c  = tid / TAPS;
        const int p  = tid % TAPS;
        const int py = p / KS, px = p % KS;
        const float xs = -1.0f + (2.0f / 6.0f) * (float)px;
        const float ys = -1.0f + (2.0f / 6.0f) * (float)py;
        const float th = (float)c * 3.14159265358979323846f / (float)NCH;
        const float ct = __cosf(th), st = __sinf(th);
        const float gx = ct * xs - st * ys;
        const float gy = st * xs + ct * ys;
        const float ix = (gx + 1.0f) * 0.5f * 6.0f;
        const float iy = (gy + 1.0f) * 0.5f * 6.0f;
        const float ix0 = floorf(ix), iy0 = floorf(iy);
        const float ix1 = ix0 + 1.0f, iy1 = iy0 + 1.0f;
        const float wx1 = ix - ix0, wx0 = 1.0f - wx1;
        const float wy1 = iy - iy0, wy0 = 1.0f - wy1;

        auto gather = [&](float iyc, float ixc) -> float {
            const bool valid = (ixc >= 0.0f) & (ixc <= 6.0f) &
                               (iyc >= 0.0f) & (iyc <= 6.0f);
            const int iyi = (int)fminf(fmaxf(iyc, 0.0f), 6.0f);
            const int ixi = (int)fminf(fmaxf(ixc, 0.0f), 6.0f);
            return valid ? kk[iyi * KS + ixi] : 0.0f;
        };

        rot[tid] = gather(iy0, ix0) * wy0 * wx0 + gather(iy0, ix1) * wy0 * wx1 +
                   gather(iy1, ix0) * wy1 * wx0 + gather(iy1, ix1) * wy1 * wx1;
    }
    __syncthreads();

    if (tid < 13 * 32) {
        const int t    = tid >> 5;
        const int lane = tid & 31;
        const int row  = lane & 15;
        const int half = lane >> 4;
        const int k0   = t * 4 + 2 * half;
        float a0 = 0.0f, a1 = 0.0f;
        if (row < NCH) {
            if (k0 < TAPS)     a0 = rot[row * TAPS + k0];
            if (k0 + 1 < TAPS) a1 = rot[row * TAPS + k0 + 1];
        }
        wprep[(t * 32 + lane) * 2 + 0] = a0;
        wprep[(t * 32 + lane) * 2 + 1] = a1;
    }
}

// ---------------------------------------------------------------------------
// Kernel 2: per block: 16 output rows x 16 cols of one image (8 waves,
// two rows per wave -> two WMMA accumulation chains, 26 WMMAs per wave).
//   - stage 22x22 input halo tile in LDS
//   - B fragments fetched as contiguous LDS pairs (ds_load_2addr 0/+1)
//     whenever both halves share the same K-step delta (9 of 13 steps)
//   - leaky-relu, exchange via LDS so EVERY lane owns one pixel
//   - 41-comparator Batcher sort of 12 channels, std (ddof=1)
//   - nontemporal (NT) output stores: 218 MB streamed, never re-read
// ---------------------------------------------------------------------------

// Batcher merge-exchange network for 16 slots with slots 12..15 elided
// (treated as +inf sentinels) -> valid 12-element sorter, 41 comparators.
__device__ __constant__ const signed char SORT_NET[41][2] = {
    {0,8},{1,9},{2,10},{3,11},
    {0,4},{1,5},{2,6},{3,7},
    {4,8},{5,9},{6,10},{7,11},
    {0,2},{1,3},{4,6},{5,7},{8,10},{9,11},
    {2,8},{3,9},
    {2,4},{3,5},{6,8},{7,9},
    {0,1},{2,3},{4,5},{6,7},{8,9},{10,11},
    {1,8},{3,10},
    {1,4},{3,6},{5,8},{7,10},
    {1,2},{3,4},{5,6},{7,8},{9,10}
};

// LDS offset of tap k within the halo tile (constant after unrolling).
__device__ __forceinline__ constexpr int offk(int k) {
    return (k < TAPS) ? ((k / KS) * TILE_STRIDE + (k % KS)) : 0;
}

__global__ __launch_bounds__(256) void conv_sort_std_kernel(
    const float* __restrict__ x, const float* __restrict__ wprep,
    float* __restrict__ out)
{
    const int bid = blockIdx.x;
    const int wt = bid & 31;         // 32 column tiles
    const int ht = (bid >> 5) & 31;  // 32 row tiles
    const int n  = bid >> 10;        // image index
    const int w0 = wt * 16;
    const int h0 = ht * 16;

    __shared__ float sin_[22 * TILE_STRIDE];  // halo tile, padded stride
    __shared__ float sbuf[8][2][16 * 17];     // per-wave x per-row exchange

    const int tid = threadIdx.x;
    const float* __restrict__ xin = x + (size_t)n * HW;

    // Cooperative halo load (zero outside the image).
    for (int idx = tid; idx < 22 * 22; idx += 256) {
        const int r  = idx / 22;
        const int cc = idx - r * 22;
        const int gh = h0 - 3 + r;
        const int gw = w0 - 3 + cc;
        float v = 0.0f;
        if ((unsigned)gh < (unsigned)IMG && (unsigned)gw < (unsigned)IMG)
            v = xin[gh * IMG + gw];
        sin_[r * TILE_STRIDE + cc] = v;
    }
    __syncthreads();

    const int lane = tid & 31;
    const int wave = tid >> 5;       // 2 output rows per wave
    const int half = lane >> 4;
    const int p    = lane & 15;      // pixel (column) within tile

    // Per-lane base into the halo tile for row chain 0 (chain 1 is +TILE_STRIDE,
    // which folds into the ds_load immediate offset).
    const int row0 = (2 * wave) * TILE_STRIDE + p;

    v8f acc0 = {};
    v8f acc1 = {};
    #pragma unroll
    for (int t = 0; t < 13; ++t) {
        const v2f a = *(const v2f*)(wprep + (t * 32 + lane) * 2);

        // Tap LDS offsets: per-t compile-time constants selected by `half`.
        const int kA0 = 4 * t,     kA1 = 4 * t + 1;   // half == 0
        const int kB0 = 4 * t + 2, kB1 = 4 * t + 3;   // half == 1
        const int dA  = offk(kA1) - offk(kA0);        // K-step delta, half == 0
        const int dB  = offk(kB1) - offk(kB0);        // K-step delta, half == 1

        const int addr = row0 + (half ? offk(kB0) : offk(kA0)); // 1 cndmask + add

        float b00 = sin_[addr];
        float b10 = sin_[addr + TILE_STRIDE];
        float b01, b11;
        if (dA == dB) {
            // Same delta for both halves -> contiguous pair: one
            // ds_load_2addr_b32 (offset0:0, offset1:dA) per chain, results
            // land in the consecutive VGPRs the WMMA B operand needs.
            b01 = sin_[addr + dA];
            b11 = sin_[addr + TILE_STRIDE + dA];
        } else {
            const int o1 = half ? offk(kB1) : offk(kA1);
            b01 = sin_[row0 + o1];
            b11 = sin_[row0 + TILE_STRIDE + o1];
        }

        // K-padding mask: dead code for t < 12, materializes only at t == 12.
        if (kA1 >= TAPS || kB1 >= TAPS) {
            const bool m0 = half ? (kB0 < TAPS) : (kA0 < TAPS);
            const bool m1 = half ? (kB1 < TAPS) : (kA1 < TAPS);
            b00 = m0 ? b00 : 0.0f;  b10 = m0 ? b10 : 0.0f;
            b01 = m1 ? b01 : 0.0f;  b11 = m1 ? b11 : 0.0f;
        }

        v2f b_r0 = { b00, b01 };
        acc0 = __builtin_amdgcn_wmma_f32_16x16x4_f32(
            false, a, false, b_r0, (short)0, acc0, false, false);

        v2f b_r1 = { b10, b11 };
        acc1 = __builtin_amdgcn_wmma_f32_16x16x4_f32(
            false, a, false, b_r1, (short)0, acc1, false, false);
    }

    // Leaky ReLU and exchange: lane p has ch 0..7, lane p+16 has ch 8..15.
    #pragma unroll
    for (int r = 0; r < 8; ++r) {
        float v0 = acc0[r];
        float v1 = acc1[r];
        v0 = (v0 > 0.0f) ? v0 : 0.01f * v0;
        v1 = (v1 > 0.0f) ? v1 : 0.01f * v1;
        sbuf[wave][0][p * 17 + r + 8 * half] = v0;
        sbuf[wave][1][p * 17 + r + 8 * half] = v1;
    }
    __syncthreads();

    // Every lane owns one pixel: row = 2*wave + (lane>>4), col = lane&15.
    float v[NCH];
    #pragma unroll
    for (int c = 0; c < NCH; ++c) v[c] = sbuf[wave][half][p * 17 + c];

    #pragma unroll
    for (int i = 0; i < 41; ++i) {
        const int a = SORT_NET[i][0], b = SORT_NET[i][1];
        const float lo = fminf(v[a], v[b]);
        const float hi = fmaxf(v[a], v[b]);
        v[a] = lo; v[b] = hi;
    }

    float s = 0.0f;
    #pragma unroll
    for (int c = 0; c < NCH; ++c) s += v[c];
    const float mean = s * (1.0f / 12.0f);
    float var = 0.0f;
    #pragma unroll
    for (int c = 0; c < NCH; ++c) { const float d = v[c] - mean; var += d * d; }
    const float sd = sqrtf(var * (1.0f / 11.0f));

    const int h = h0 + 2 * wave + half;
    const int w = w0 + p;
    const size_t base = (size_t)n * 13 * HW + (size_t)h * IMG + w;
    #pragma unroll
    for (int c = 0; c < NCH; ++c)
        __builtin_nontemporal_store(v[c], &out[base + (size_t)c * HW]);
    __builtin_nontemporal_store(sd, &out[base + (size_t)NCH * HW]);
}

extern "C" void kernel_launch(void* const* d_in, const int* in_sizes, int n_in,
                              void* d_out, int out_size, void* d_ws, size_t ws_size,
                              hipStream_t stream)
{
    const float* x    = (const float*)d_in[0];   // (16,1,512,512) f32
    const float* k2d  = (const float*)d_in[1];   // (1,1,7,7) f32
    float* out        = (float*)d_out;           // (16,13,512,512) f32
    float* wprep      = (float*)d_ws;            // 13*32*2 floats = 3328 B

    rot_prep_kernel<<<1, 640, 0, stream>>>(k2d, wprep);

    const int blocks = 16 * 32 * 32;             // images * row-tiles * col-tiles
    conv_sort_std_kernel<<<blocks, 256, 0, stream>>>(x, wprep, out);
}